// MedianFilter_39281770889998
// MI455X (gfx1250) — compile-verified
//
#include <hip/hip_runtime.h>
#include <hip/hip_bf16.h>

#ifndef __has_builtin
#define __has_builtin(x) 0
#endif

#if __has_builtin(__builtin_amdgcn_global_load_async_to_lds_b32)
#define HAVE_ASYNC_B32 1
#else
#define HAVE_ASYNC_B32 0
#endif

typedef __attribute__((address_space(1))) int* gas_i32;   // global int*
typedef __attribute__((address_space(3))) int* las_i32;   // LDS int*

namespace {
constexpr int H = 512, W = 512, BC = 16 * 3;
constexpr int TILE_W = 128, TILE_H = 8;
constexpr int LDSW = TILE_W + 2;   // 130
constexpr int LDSH = TILE_H + 2;   // 10
constexpr int NLDS = LDSW * LDSH;  // 1300
constexpr int NTHREADS = 256;
}

__device__ __forceinline__ float med3f(float a, float b, float c) {
  return __builtin_amdgcn_fmed3f(a, b, c);   // v_med3_num_f32
}
__device__ __forceinline__ float min3f(float a, float b, float c) {
  return fminf(fminf(a, b), c);               // fuses to v_min3_num_f32
}
__device__ __forceinline__ float max3f(float a, float b, float c) {
  return fmaxf(fmaxf(a, b), c);               // fuses to v_max3_num_f32
}

__global__ __launch_bounds__(NTHREADS)
void median3x3_kernel(const float* __restrict__ in, float* __restrict__ out) {
  __shared__ float tile[NLDS];

  const int tid = threadIdx.x;
  const int x0 = blockIdx.x * TILE_W;
  const int y0 = blockIdx.y * TILE_H;
  const int bc = blockIdx.z;
  const float* gin = in + (size_t)bc * (H * W);

  // ---- Stage (TILE_H+2) x (TILE_W+2) halo tile into LDS (zero-padded) ----
  for (int i = tid; i < NLDS; i += NTHREADS) {
    const int ly = i / LDSW;
    const int lx = i - ly * LDSW;
    const int gy = y0 - 1 + ly;
    const int gx = x0 - 1 + lx;
    const bool inb = ((unsigned)gy < (unsigned)H) && ((unsigned)gx < (unsigned)W);
#if HAVE_ASYNC_B32
    if (inb) {
      const float* gp = gin + gy * W + gx;
      // CDNA5 async DMA: global -> LDS, tracked by ASYNCcnt
      __builtin_amdgcn_global_load_async_to_lds_b32(
          (gas_i32)gp, (las_i32)&tile[i], 0, 0);
    } else {
      tile[i] = 0.0f;   // zero padding written directly to LDS
    }
#else
    tile[i] = inb ? gin[gy * W + gx] : 0.0f;
#endif
  }

#if HAVE_ASYNC_B32
#if __has_builtin(__builtin_amdgcn_s_wait_asynccnt)
  __builtin_amdgcn_s_wait_asynccnt(0);
#else
  asm volatile("s_wait_asynccnt 0" ::: "memory");
#endif
#endif
  __syncthreads();

  // ---- Median-of-9 via med3 network; 4 consecutive outputs per thread ----
  const int tx = tid & 31;        // 32 column-groups
  const int ty = tid >> 5;        // 8 rows
  const int base = tx << 2;       // first of 4 output columns (tile-local)

  float lo[6], mi[6], hi[6];
#pragma unroll
  for (int j = 0; j < 6; ++j) {
    const float a = tile[(ty + 0) * LDSW + base + j];
    const float b = tile[(ty + 1) * LDSW + base + j];
    const float c = tile[(ty + 2) * LDSW + base + j];
    lo[j] = min3f(a, b, c);
    mi[j] = med3f(a, b, c);
    hi[j] = max3f(a, b, c);
  }

  float r[4];
#pragma unroll
  for (int k = 0; k < 4; ++k) {
    r[k] = med3f(max3f(lo[k], lo[k + 1], lo[k + 2]),
                 med3f(mi[k], mi[k + 1], mi[k + 2]),
                 min3f(hi[k], hi[k + 1], hi[k + 2]));
  }

  float4 o;
  o.x = r[0]; o.y = r[1]; o.z = r[2]; o.w = r[3];

  const int gy = y0 + ty;
  const int gx = x0 + base;
  float* op = out + (size_t)bc * (H * W) + gy * W + gx;  // 16B aligned
  *reinterpret_cast<float4*>(op) = o;                     // global_store_b128
}

extern "C" void kernel_launch(void* const* d_in, const int* in_sizes, int n_in,
                              void* d_out, int out_size, void* d_ws, size_t ws_size,
                              hipStream_t stream) {
  const float* x = (const float*)d_in[0];
  float* out = (float*)d_out;
  dim3 grid(W / TILE_W, H / TILE_H, BC);   // 4 x 64 x 48
  dim3 block(NTHREADS);
  hipLaunchKernelGGL(median3x3_kernel, grid, block, 0, stream, x, out);
}